// CodaPrompt_1563368096221
// MI455X (gfx1250) — compile-verified
//
#include <hip/hip_runtime.h>
#include <math.h>

// ---------------- problem constants (from reference) ----------------
#define POOLK   20          // F_END = POOL/N_TASKS
#define TOPKN   10
#define KEYD    768
#define EPLEN   8
#define PFEAT   9216        // EMB_D * HEADS
#define NCOLS   (EPLEN * PFEAT)   // 73728 = flattened (l,d) columns of ps
#define NROWS   1024              // B*Q = 4*256
#define HALF_OUT ((size_t)NROWS * 4 * PFEAT)   // Ek size = 37,748,736 elems

typedef float v2f __attribute__((ext_vector_type(2)));
typedef float v8f __attribute__((ext_vector_type(8)));

// =====================================================================
// Kernel 1: scores -> top-10 -> softmax -> dense gate row (1024 x 20)
// One wave32 per (b,q) row. x row preloaded into 24 VGPRs/lane and
// reused for all 20 prompts; K/A rows stream from L2 (60 KB hot set).
// =====================================================================
__global__ __launch_bounds__(256) void gate_kernel(
    const float* __restrict__ x,   // [1024][768]
    const float* __restrict__ K,   // [100][768] (use first 20 rows)
    const float* __restrict__ A,   // [100][768]
    float* __restrict__ gate)      // [1024][20] (workspace)
{
  const int wave = (blockIdx.x * blockDim.x + threadIdx.x) >> 5;
  const int lane = threadIdx.x & 31;
  if (wave >= NROWS) return;

  const float* xr = x + (size_t)wave * KEYD;
  float xv[KEYD / 32];
  #pragma unroll
  for (int i = 0; i < KEYD / 32; ++i) xv[i] = xr[lane + 32 * i];

  float sc[POOLK];
  for (int k = 0; k < POOLK; ++k) {
    const float* Kr = K + (size_t)k * KEYD;
    const float* Ar = A + (size_t)k * KEYD;
    float num = 0.f, den = 0.f, kn = 0.f;
    #pragma unroll
    for (int i = 0; i < KEYD / 32; ++i) {
      const int d = lane + 32 * i;
      const float kv = Kr[d];
      const float xa = xv[i] * Ar[d];
      num = fmaf(xa, kv, num);
      den = fmaf(xa, xa, den);
      kn  = fmaf(kv, kv, kn);
    }
    // wave32 butterfly reduction
    #pragma unroll
    for (int m = 16; m >= 1; m >>= 1) {
      num += __shfl_xor(num, m, 32);
      den += __shfl_xor(den, m, 32);
      kn  += __shfl_xor(kn,  m, 32);
    }
    const float dn = fmaxf(sqrtf(den), 1e-12f) * fmaxf(sqrtf(kn), 1e-12f);
    sc[k] = num / dn;
  }

  if (lane == 0) {
    // top-10 selection; strict '>' => earliest index wins ties (jax top_k)
    int   sel[TOPKN];
    float val[TOPKN];
    unsigned used = 0u;
    for (int t = 0; t < TOPKN; ++t) {
      int best = 0; float bv = -3.402823e38f;
      for (int k = 0; k < POOLK; ++k)
        if (!(used & (1u << k)) && sc[k] > bv) { bv = sc[k]; best = k; }
      used |= 1u << best;
      sel[t] = best; val[t] = bv;
    }
    // softmax over the 10 kept values (val[0] is the max)
    const float mx = val[0];
    float ssum = 0.f;
    #pragma unroll
    for (int t = 0; t < TOPKN; ++t) { val[t] = expf(val[t] - mx); ssum += val[t]; }
    const float inv = 1.f / ssum;

    float g[POOLK];
    #pragma unroll
    for (int k = 0; k < POOLK; ++k) g[k] = 0.f;
    #pragma unroll
    for (int t = 0; t < TOPKN; ++t) g[sel[t]] = val[t] * inv;

    float* gr = gate + (size_t)wave * POOLK;
    #pragma unroll
    for (int k = 0; k < POOLK; ++k) gr[k] = g[k];
  }
}

// =====================================================================
// Kernel 2: out[1024 x 73728] = gate[1024 x 20] * ps[20 x 73728]
// via V_WMMA_F32_16X16X4_F32 (K=20 = 5 k-steps). Each wave owns a
// 16-row M-tile x 128-col strip (8 subtiles of 16x16).
//
// All loads/stores hang off TWO per-wave base pointers with
// compile-time immediate offsets (24-bit IOFFSET):
//   B load : s*4*NCOLS*4  <= 4.72 MB, + NCOLS*4, + sub*64 B  (< 8.38 MB)
//   D store: j*4*PFEAT*4  <= 1.03 MB, + sub*64 B
// A 128-col strip never crosses an l boundary (9216 % 128 == 0), so the
// Ek/Ev split is resolved once per wave.
// =====================================================================
#define NSUB 8                              // 16-col subtiles per wave
#define NSTRIPS (NCOLS / (16 * NSUB))       // 576

__global__ __launch_bounds__(256) void gemm_kernel(
    const float* __restrict__ gate,  // [1024][20]
    const float* __restrict__ p,     // rows 0..19 used, flat [k][73728]
    float* __restrict__ out)         // Ek (1024*4*9216) then Ev (...)
{
  const int waveInBlk = threadIdx.x >> 5;          // 0..7
  const int lane      = threadIdx.x & 31;
  const int strip     = blockIdx.x * 8 + waveInBlk; // 0..575 (72 blocks * 8)
  const int mTile     = blockIdx.y;                 // 0..63

  const int half = lane >> 4;               // 0: lanes 0-15, 1: lanes 16-31
  const int l16  = lane & 15;

  // --- A tile (16x4 f32 per k-step): v0 holds K=+0/+2, v1 holds K=+1/+3
  v2f a[5];
  {
    const float* gr = gate + (size_t)(mTile * 16 + l16) * POOLK + 2 * half;
    #pragma unroll
    for (int s = 0; s < 5; ++s) {
      a[s][0] = gr[4 * s];
      a[s][1] = gr[4 * s + 1];
    }
  }

  // --- per-wave bases (the only 64-bit address math in the kernel)
  const int colStrip = strip * (16 * NSUB);        // 128-aligned
  const int l     = colStrip / PFEAT;              // constant over the strip
  const int dBase = colStrip % PFEAT;

  const float* pb = p + (size_t)(2 * half) * NCOLS + colStrip + l16;
  float* ob = out + ((l < 4) ? (size_t)0 : HALF_OUT)
                  + ((size_t)(mTile * 16 + 8 * half) * 4 + (size_t)(l & 3)) * PFEAT
                  + dBase + l16;

  #pragma unroll
  for (int sub = 0; sub < NSUB; ++sub) {
    v8f c = {0.f, 0.f, 0.f, 0.f, 0.f, 0.f, 0.f, 0.f};
    #pragma unroll
    for (int s = 0; s < 5; ++s) {
      v2f b;                                       // B tile (4x16 f32)
      b[0] = pb[sub * 16 + s * 4 * NCOLS];         // row K = 4s + 2*half
      b[1] = pb[sub * 16 + s * 4 * NCOLS + NCOLS]; // row K = 4s + 2*half + 1
      c = __builtin_amdgcn_wmma_f32_16x16x4_f32(
              /*neg_a=*/false, a[s], /*neg_b=*/false, b,
              /*c_mod=*/(short)0, c, /*reuse_a=*/false, /*reuse_b=*/false);
    }
    // --- 16x16 C/D tile: VGPR j -> rows j / j+8; immediate-offset NT stores
    #pragma unroll
    for (int j = 0; j < 8; ++j)
      __builtin_nontemporal_store(c[j], ob + sub * 16 + (size_t)j * 4 * PFEAT);
  }
}

// =====================================================================
extern "C" void kernel_launch(void* const* d_in, const int* in_sizes, int n_in,
                              void* d_out, int out_size, void* d_ws, size_t ws_size,
                              hipStream_t stream) {
  const float* x = (const float*)d_in[0];   // x_querry [4,256,768]
  // d_in[1] = l (unused: F_END is a compile-time constant in the reference)
  // d_in[2] = x_block (unused by the reference)
  const float* K = (const float*)d_in[3];   // [100,768]
  const float* A = (const float*)d_in[4];   // [100,768]
  const float* p = (const float*)d_in[5];   // [100,8,9216]
  float* out = (float*)d_out;
  float* gate = (float*)d_ws;               // 1024*20 floats = 80 KB scratch

  // 1024 waves, 8 waves / 256-thread block
  gate_kernel<<<NROWS / 8, 256, 0, stream>>>(x, K, A, gate);

  // grid: x = 576 strips / 8 waves-per-block, y = 64 M-tiles
  gemm_kernel<<<dim3(NSTRIPS / 8, NROWS / 16), 256, 0, stream>>>(gate, p, out);
}